// so3cnn_equators_16475494547638
// MI455X (gfx1250) — compile-verified
//
#include <hip/hip_runtime.h>

typedef __attribute__((ext_vector_type(2))) float v2f;
typedef __attribute__((ext_vector_type(8))) float v8f;
typedef __attribute__((ext_vector_type(4))) unsigned int u32x4;
typedef __attribute__((ext_vector_type(8))) int i32x8;
typedef __attribute__((ext_vector_type(4))) int i32x4;

#define PI_F 3.14159265358979323846f

#ifndef __has_builtin
#define __has_builtin(x) 0
#endif
#if __has_builtin(__builtin_amdgcn_tensor_load_to_lds) && \
    __has_builtin(__builtin_amdgcn_s_wait_tensorcnt)
#define HAVE_TDM 1
#else
#define HAVE_TDM 0
#endif

static __device__ __forceinline__ v8f wmma_f32(v2f a, v2f b, v8f c) {
  // D = A(16x4,f32) * B(4x16,f32) + C(16x16,f32)
  return __builtin_amdgcn_wmma_f32_16x16x4_f32(false, a, false, b, (short)0, c, false, false);
}

#if HAVE_TDM
// TDM: DMA n8 8-byte elements from global 'g' into LDS offset lds_off.
// D# per CDNA5 ISA ch.8: group0 {count=1, lds_addr, global_addr, type=2},
// group1 {data_size=8B, tensor_dim0=n8, tensor_dim1=1, tile_dim0=n8, tile_dim1=1,
//         tensor_dim0_stride=n8}; groups 2/3 unused (<=2D tensor).
static __device__ __forceinline__ void tdm_load_1d(unsigned lds_off, const void* g, int n8) {
  unsigned long long ga = (unsigned long long)g;
  u32x4 g0 = {};
  g0[0] = 1u;                                              // count = 1
  g0[1] = lds_off;                                         // lds_addr
  g0[2] = (unsigned)ga;                                    // global_addr[31:0]
  g0[3] = (unsigned)((ga >> 32) & 0x1ffffffull) | (2u << 30);  // ga[56:32], type=2
  i32x8 g1 = {};
  g1[0] = 3 << 16;                                         // data_size = 3 (8 bytes)
  g1[1] = (n8 & 0xffff) << 16;                             // tensor_dim0[15:0]
  g1[2] = ((n8 >> 16) & 0xffff) | (1 << 16);               // dim0[31:16], tensor_dim1=1
  g1[3] = (n8 & 0xffff) << 16;                             // tile_dim0 = n8
  g1[4] = 1;                                               // tile_dim1 = 1
  g1[5] = n8;                                              // tensor_dim0_stride
  i32x4 gz = {};
#if __clang_major__ >= 23
  i32x8 gz8 = {};
  __builtin_amdgcn_tensor_load_to_lds(g0, g1, gz, gz, gz8, 0);
#else
  __builtin_amdgcn_tensor_load_to_lds(g0, g1, gz, gz, 0);
#endif
  __builtin_amdgcn_s_wait_tensorcnt(0);
}
#endif

// ---------------------------------------------------------------------------
// Setup kernels (tables recomputed every call; deterministic, capture-safe)
// ---------------------------------------------------------------------------

__global__ void k_lf(float* lf, int nmax) {
  if (blockIdx.x == 0 && threadIdx.x == 0) {
    double a = 0.0;
    lf[0] = 0.f;
    for (int i = 1; i <= nmax; ++i) { a += log((double)i); lf[i] = (float)a; }
  }
}

__global__ void k_wq(float* wq, int b) {
  int j = blockIdx.x * blockDim.x + threadIdx.x;
  int n = 2 * b;
  if (j >= n) return;
  float acc = 0.f;
  for (int k = 0; k < b; ++k)
    acc += sinf((float)(2 * j + 1) * (float)(2 * k + 1) * PI_F / (4.f * (float)b)) /
           (float)(2 * k + 1);
  float w = 2.f / (float)b * sinf(PI_F * (float)(2 * j + 1) / (4.f * (float)b)) * acc;
  wq[j] = w / (2.f * (2.f * b) * (2.f * b));
}

// Twiddle table: W[k*n+j] = exp(sign * i * 2*pi*k*j/n)
__global__ void k_twiddle(float2* W, int n, float sign) {
  int t = blockIdx.x * blockDim.x + threadIdx.x;
  if (t >= n * n) return;
  int k = t / n, j = t % n;
  float th = sign * (2.f * PI_F) * (float)((k * j) % n) / (float)n;
  float s, c;
  __sincosf(th, &s, &c);
  W[t] = make_float2(c, s);
}

// Wigner little-d tables: dst[((l*nbeta+z)*32+mi)*32+ni], mi,ni in [0,2l], else 0.
// wmode: 0 -> * wq[z], 1 -> * (2l+1), 2 -> * 1
__global__ void k_wigner(float* dst, const float* __restrict__ lf,
                         const float* __restrict__ wq, int lcount, int nbeta,
                         int pi2mode, int ndiv, int wmode) {
  size_t t = (size_t)blockIdx.x * blockDim.x + threadIdx.x;
  size_t total = (size_t)lcount * nbeta * 32 * 32;
  if (t >= total) return;
  int ni = (int)(t & 31);
  int mi = (int)((t >> 5) & 31);
  size_t rem = t >> 10;
  int z = (int)(rem % nbeta);
  int l = (int)(rem / nbeta);
  float val = 0.f;
  if (mi <= 2 * l && ni <= 2 * l) {
    int mu = mi - l, nu = ni - l;
    float beta = pi2mode ? (0.5f * PI_F) : ((z + 0.5f) * PI_F / (float)ndiv);
    float cb = cosf(0.5f * beta), sb = sinf(0.5f * beta);
    float lognum = 0.5f * (lf[l + mu] + lf[l - mu] + lf[l + nu] + lf[l - nu]);
    for (int s = 0; s <= 2 * l; ++s) {
      if (l + nu - s >= 0 && mu - nu + s >= 0 && l - mu - s >= 0) {
        float logden = lf[l + nu - s] + lf[s] + lf[mu - nu + s] + lf[l - mu - s];
        float coef = expf(lognum - logden);
        if ((mu - nu + s) & 1) coef = -coef;
        int ec = 2 * l + nu - mu - 2 * s;
        int es = mu - nu + 2 * s;
        val += coef * powf(cb, (float)ec) * powf(sb, (float)es);
      }
    }
    float w = (wmode == 0) ? wq[z] : ((wmode == 1) ? (float)(2 * l + 1) : 1.f);
    val *= w;
  }
  dst[t] = val;
}

// khat[io][d] = scal * sum_g kern[io][g] * exp(+i * d * 2*pi*g/ng)
__global__ void k_khat(const float* __restrict__ kern, float2* __restrict__ khat,
                       int nio, int ng, float scal) {
  int t = blockIdx.x * blockDim.x + threadIdx.x;
  if (t >= nio * ng) return;
  int d = t % ng, io = t / ng;
  float sr = 0.f, si = 0.f;
  for (int g = 0; g < ng; ++g) {
    float th = (2.f * PI_F) * (float)((d * g) % ng) / (float)ng;
    float s, c;
    __sincosf(th, &s, &c);
    float kv = kern[io * ng + g];
    sr += kv * c;
    si += kv * s;
  }
  khat[t] = make_float2(sr * scal, si * scal);
}

__global__ void k_fillz(float* p, size_t n) {
  size_t t = (size_t)blockIdx.x * blockDim.x + threadIdx.x;
  if (t < n) p[t] = 0.f;
}

// ---------------------------------------------------------------------------
// WMMA DFT: Y[r,j] = sum_k X[r,k] * W[k,j], W staged into LDS (TDM if present).
// Strided addressing: elem(r,k) = (r/inner)*outer + (r%inner) + k*kstride.
// One wave per 16-row strip; computes all N/16 column tiles (A reused).
// A frag (16x4 f32): lanes 0-15 rows {K0,K1}; lanes 16-31 rows {K2,K3}
// B frag (4x16 f32): lanes 0-15 cols rows {K0,K1}; lanes 16-31 {K2,K3}
// C/D (16x16 f32, 8 VGPR): vgpr rr -> M=rr (+8 for lanes 16-31), N=lane&15
// ---------------------------------------------------------------------------
template <int N, int REAL, int RELU>
__global__ __launch_bounds__(32) void k_dft_t(const float2* __restrict__ Xc,
                                              const float* __restrict__ Xr,
                                              float2* __restrict__ Y,
                                              const float2* __restrict__ Wtab,
                                              int inner, int outer, int kstride) {
  constexpr int NT = N / 16;
  __shared__ float2 w_lds[N * N];
#if HAVE_TDM
  tdm_load_1d((unsigned)(size_t)(void*)w_lds, (const void*)Wtab, N * N);
#else
  for (int idx = threadIdx.x; idx < N * N; idx += 32) w_lds[idx] = Wtab[idx];
#endif
  __syncthreads();

  int tr = blockIdx.x;
  int lane = threadIdx.x;
  int half = lane >> 4;
  int l15 = lane & 15;
  int r = tr * 16 + l15;
  size_t xbase = (size_t)(r / inner) * outer + (size_t)(r % inner);

  v8f cr[NT], ci[NT];
#pragma unroll
  for (int t = 0; t < NT; ++t) { cr[t] = {}; ci[t] = {}; }

  for (int k0 = 0; k0 < N; k0 += 4) {
    int kb = k0 + 2 * half;
    v2f ar, ai, nai;
    if (REAL) {
      float x0 = Xr[xbase + (size_t)kb * kstride];
      float x1 = Xr[xbase + (size_t)(kb + 1) * kstride];
      if (RELU) { x0 = fmaxf(x0, 0.f); x1 = fmaxf(x1, 0.f); }
      ar.x = x0; ar.y = x1;
      ai.x = 0.f; ai.y = 0.f;
      nai = ai;
    } else {
      __builtin_prefetch(Xc + xbase + (size_t)(kb + 8) * kstride, 0, 1);
      float2 x0 = Xc[xbase + (size_t)kb * kstride];
      float2 x1 = Xc[xbase + (size_t)(kb + 1) * kstride];
      ar.x = x0.x; ar.y = x1.x;
      ai.x = x0.y; ai.y = x1.y;
      nai.x = -x0.y; nai.y = -x1.y;
    }
#pragma unroll
    for (int t = 0; t < NT; ++t) {
      int j = t * 16 + l15;
      float2 w0 = w_lds[kb * N + j];
      float2 w1 = w_lds[(kb + 1) * N + j];
      v2f br, bi;
      br.x = w0.x; br.y = w1.x;
      bi.x = w0.y; bi.y = w1.y;
      cr[t] = wmma_f32(ar, br, cr[t]);
      ci[t] = wmma_f32(ar, bi, ci[t]);
      if (!REAL) {
        cr[t] = wmma_f32(nai, bi, cr[t]);  // Cr += -Ai*Bi
        ci[t] = wmma_f32(ai, br, ci[t]);   // Ci += Ai*Br
      }
    }
  }
#pragma unroll
  for (int t = 0; t < NT; ++t) {
    int j = t * 16 + l15;
#pragma unroll
    for (int rr = 0; rr < 8; ++rr) {
      int ro = tr * 16 + rr + 8 * half;
      size_t oi =
          (size_t)(ro / inner) * outer + (size_t)(ro % inner) + (size_t)j * kstride;
      Y[oi] = make_float2(cr[t][rr], ci[t][rr]);
    }
  }
}

// ---------------------------------------------------------------------------
// WMMA complex GEMM: C[M,N] = sum_k A[M,K]*B[K,N] (row-major, padded, K%4==0)
// ---------------------------------------------------------------------------
__global__ __launch_bounds__(32) void k_zgemm(const float2* __restrict__ A,
                                              const float2* __restrict__ B,
                                              float2* __restrict__ C, int Ntiles, int K,
                                              int lda, int ldb, int ldc) {
  int tile = blockIdx.x;
  int tr = tile / Ntiles, tc = tile % Ntiles;
  int lane = threadIdx.x;
  int half = lane >> 4;
  int l15 = lane & 15;
  int r = tr * 16 + l15;
  int j = tc * 16 + l15;
  v8f cr = {};
  v8f ci = {};
  for (int k0 = 0; k0 < K; k0 += 4) {
    int kb = k0 + 2 * half;
    float2 a0 = A[(size_t)r * lda + kb];
    float2 a1 = A[(size_t)r * lda + kb + 1];
    float2 b0 = B[(size_t)kb * ldb + j];
    float2 b1 = B[(size_t)(kb + 1) * ldb + j];
    v2f ar, ai, nai, br, bi;
    ar.x = a0.x; ar.y = a1.x;
    ai.x = a0.y; ai.y = a1.y;
    nai.x = -a0.y; nai.y = -a1.y;
    br.x = b0.x; br.y = b1.x;
    bi.x = b0.y; bi.y = b1.y;
    cr = wmma_f32(ar, br, cr);
    cr = wmma_f32(nai, bi, cr);
    ci = wmma_f32(ar, bi, ci);
    ci = wmma_f32(ai, br, ci);
  }
  for (int rr = 0; rr < 8; ++rr) {
    int ro = tr * 16 + rr + 8 * half;
    C[(size_t)ro * ldc + j] = make_float2(cr[rr], ci[rr]);
  }
}

// ---------------------------------------------------------------------------
// Per-degree small kernels
// ---------------------------------------------------------------------------

// A[(m*Bn+b)][(k*fi+i)] = sum_z xhat[b,i,z, mod(m-l,n), mod(k-l,n)] * din[l,z,m,k]
__global__ void k_fx(const float2* __restrict__ xhat, const float* __restrict__ din,
                     float2* __restrict__ A, int l, int Bn, int fi, int nz, int n,
                     int lda) {
  int tl = 2 * l + 1;
  size_t total = (size_t)tl * Bn * tl * fi;
  size_t t = (size_t)blockIdx.x * blockDim.x + threadIdx.x;
  if (t >= total) return;
  int i = (int)(t % fi);
  size_t r1 = t / fi;
  int k = (int)(r1 % tl);
  size_t r2 = r1 / tl;
  int b = (int)(r2 % Bn);
  int m = (int)(r2 / Bn);
  int ma = ((m - l) % n + n) % n;
  int mg = ((k - l) % n + n) % n;
  const float2* xp = xhat + ((size_t)(b * fi + i) * nz) * n * n + (size_t)ma * n + mg;
  const float* dp = din + (((size_t)l * nz) * 32 + m) * 32 + k;
  float2 acc = make_float2(0.f, 0.f);
  for (int z = 0; z < nz; ++z) {
    float2 xv = xp[(size_t)z * n * n];
    float dv = dp[(size_t)z * 1024];
    acc.x += xv.x * dv;
    acc.y += xv.y * dv;
  }
  A[(size_t)(m * Bn + b) * lda + (k * fi + i)] = acc;
}

// Bm[(k*fi+i)][(n*fo+o)] = dpi2[l,n,k] * conj(khat[i,o, mod(k-n, ng)])
__global__ void k_fyb(const float2* __restrict__ khat, const float* __restrict__ dpi2,
                      float2* __restrict__ Bm, int l, int fi, int fo, int ng, int ldb) {
  int tl = 2 * l + 1;
  size_t total = (size_t)tl * fi * tl * fo;
  size_t t = (size_t)blockIdx.x * blockDim.x + threadIdx.x;
  if (t >= total) return;
  int o = (int)(t % fo);
  size_t r1 = t / fo;
  int nn = (int)(r1 % tl);
  size_t r2 = r1 / tl;
  int i = (int)(r2 % fi);
  int k = (int)(r2 / fi);
  int d = ((k - nn) % ng + ng) % ng;
  float2 kh = khat[(size_t)(i * fo + o) * ng + d];
  float dp = dpi2[((size_t)l * 32 + nn) * 32 + k];
  Bm[(size_t)(k * fi + i) * ldb + (nn * fo + o)] = make_float2(dp * kh.x, -dp * kh.y);
}

// S[b,o,z, mod(m-l,no), mod(nn-l,no)] += C[(m*Bn+b)][(nn*fo+o)] * dout[l,z,m,nn]
__global__ void k_contrib(const float2* __restrict__ C, const float* __restrict__ dout,
                          float2* __restrict__ S, int l, int Bn, int fo, int nz,
                          int no, int ldc) {
  int tl = 2 * l + 1;
  size_t total = (size_t)Bn * fo * tl * tl;
  size_t t = (size_t)blockIdx.x * blockDim.x + threadIdx.x;
  if (t >= total) return;
  int nn = (int)(t % tl);
  size_t r1 = t / tl;
  int m = (int)(r1 % tl);
  size_t r2 = r1 / tl;
  int o = (int)(r2 % fo);
  int b = (int)(r2 / fo);
  float2 fz = C[(size_t)(m * Bn + b) * ldc + (nn * fo + o)];
  int ma = ((m - l) % no + no) % no;
  int mg = ((nn - l) % no + no) % no;
  float2* sp = S + ((size_t)(b * fo + o) * nz) * no * no + (size_t)ma * no + mg;
  const float* dp = dout + (((size_t)l * nz) * 32 + m) * 32 + nn;
  for (int z = 0; z < nz; ++z) {
    float dv = dp[(size_t)z * 1024];
    float2 cur = sp[(size_t)z * no * no];
    cur.x += fz.x * dv;
    cur.y += fz.y * dv;
    sp[(size_t)z * no * no] = cur;
  }
}

__global__ void k_final(const float2* __restrict__ Z, const float* __restrict__ bias,
                        float* __restrict__ out, size_t total, int per_o, int fo) {
  size_t t = (size_t)blockIdx.x * blockDim.x + threadIdx.x;
  if (t >= total) return;
  int o = (int)((t / per_o) % fo);
  out[t] = Z[t].x + bias[o];
}

// ---------------------------------------------------------------------------
// Host orchestration
// ---------------------------------------------------------------------------
static inline unsigned nblk(size_t n) { return (unsigned)((n + 255) / 256); }

extern "C" void kernel_launch(void* const* d_in, const int* in_sizes, int n_in,
                              void* d_out, int out_size, void* d_ws, size_t ws_size,
                              hipStream_t stream) {
  (void)in_sizes; (void)n_in; (void)out_size; (void)ws_size;
  const float* x = (const float*)d_in[0];        // [8,4,64,64,64]
  const float* kernel1 = (const float*)d_in[1];  // [4,24,64]
  const float* bias1 = (const float*)d_in[2];    // [24]
  const float* kernel2 = (const float*)d_in[3];  // [24,32,32]
  const float* bias2 = (const float*)d_in[4];    // [32]
  float* out = (float*)d_out;                    // x1 (6291456) ++ x2 (1048576)

  // ---- workspace carve-up ----
  char* ws = (char*)d_ws;
  size_t off = 0;
  auto carve = [&](size_t bytes) -> char* {
    char* p = ws + off;
    off += (bytes + 255) & ~(size_t)255;
    return p;
  };
  float* lf = (float*)carve(65 * sizeof(float));
  float* wq1 = (float*)carve(64 * sizeof(float));
  float* wq2 = (float*)carve(32 * sizeof(float));
  float* din1 = (float*)carve((size_t)16 * 64 * 1024 * sizeof(float));
  float* dout1 = (float*)carve((size_t)16 * 32 * 1024 * sizeof(float));
  float* dpi1 = (float*)carve((size_t)16 * 1024 * sizeof(float));
  float* din2 = (float*)carve((size_t)8 * 32 * 1024 * sizeof(float));
  float* dout2 = (float*)carve((size_t)8 * 16 * 1024 * sizeof(float));
  float* dpi2t = (float*)carve((size_t)8 * 1024 * sizeof(float));
  float2* khat1 = (float2*)carve((size_t)96 * 64 * sizeof(float2));
  float2* khat2 = (float2*)carve((size_t)768 * 32 * sizeof(float2));
  float2* W64f = (float2*)carve((size_t)64 * 64 * sizeof(float2));
  float2* W32f = (float2*)carve((size_t)32 * 32 * sizeof(float2));
  float2* W32i = (float2*)carve((size_t)32 * 32 * sizeof(float2));
  float2* W16i = (float2*)carve((size_t)16 * 16 * sizeof(float2));
  float2* bufA = (float2*)carve((size_t)8388608 * sizeof(float2));  // 64 MiB
  float2* bufB = (float2*)carve((size_t)8388608 * sizeof(float2));  // 64 MiB
  float2* Afz = (float2*)carve((size_t)256 * 384 * sizeof(float2));
  float2* Bfz = (float2*)carve((size_t)384 * 768 * sizeof(float2));
  float2* Cfz = (float2*)carve((size_t)256 * 768 * sizeof(float2));
  const int LDA = 384, LDB = 768, LDC = 768;

  // ---- tables ----
  k_lf<<<1, 1, 0, stream>>>(lf, 64);
  k_wq<<<1, 64, 0, stream>>>(wq1, 32);
  k_wq<<<1, 32, 0, stream>>>(wq2, 16);
  k_twiddle<<<nblk(64 * 64), 256, 0, stream>>>(W64f, 64, -1.f);
  k_twiddle<<<nblk(32 * 32), 256, 0, stream>>>(W32f, 32, -1.f);
  k_twiddle<<<nblk(32 * 32), 256, 0, stream>>>(W32i, 32, 1.f);
  k_twiddle<<<nblk(16 * 16), 256, 0, stream>>>(W16i, 16, 1.f);
  k_wigner<<<nblk((size_t)16 * 64 * 1024), 256, 0, stream>>>(din1, lf, wq1, 16, 64, 0, 64, 0);
  k_wigner<<<nblk((size_t)16 * 32 * 1024), 256, 0, stream>>>(dout1, lf, wq1, 16, 32, 0, 32, 1);
  k_wigner<<<nblk((size_t)16 * 1024), 256, 0, stream>>>(dpi1, lf, wq1, 16, 1, 1, 1, 2);
  k_wigner<<<nblk((size_t)8 * 32 * 1024), 256, 0, stream>>>(din2, lf, wq2, 8, 32, 0, 32, 0);
  k_wigner<<<nblk((size_t)8 * 16 * 1024), 256, 0, stream>>>(dout2, lf, wq2, 8, 16, 0, 16, 1);
  k_wigner<<<nblk((size_t)8 * 1024), 256, 0, stream>>>(dpi2t, lf, wq2, 8, 1, 1, 1, 2);
  const float scal1 = 0.1767766952966368811f;  // 1/sqrt(32)
  const float scal2 = 0.1020620726159657527f;  // 1/sqrt(96)
  k_khat<<<nblk(96 * 64), 256, 0, stream>>>(kernel1, khat1, 96, 64, scal1);
  k_khat<<<nblk(768 * 32), 256, 0, stream>>>(kernel2, khat2, 768, 32, scal2);

  // ============================ Layer 1 ============================
  // b_in=32 (n=64, nz=64), b_out=16 (no=32, nzo=32), fi=4, fo=24, ng=64
  {
    const int Bn = 8, FI = 4, FO = 24, N = 64, NZ = 64, NO = 32, NZO = 32, NG = 64;
    int R = Bn * FI * NZ * N;  // 131072 rows for both axes
    // forward DFT along gamma (real input), then along alpha (complex, strided)
    k_dft_t<64, 1, 0><<<R / 16, 32, 0, stream>>>(nullptr, x, bufA, W64f, 1, N, 1);
    k_dft_t<64, 0, 0><<<R / 16, 32, 0, stream>>>(bufA, nullptr, bufB, W64f, N, N * N, N);
    // S accumulator = bufA (zeroed)
    size_t sflt = (size_t)2 * Bn * FO * NZO * NO * NO;
    k_fillz<<<nblk(sflt), 256, 0, stream>>>((float*)bufA, sflt);
    for (int l = 0; l < 16; ++l) {
      int tl = 2 * l + 1;
      int Mp = ((tl * Bn + 15) / 16) * 16;
      int Np = ((tl * FO + 15) / 16) * 16;
      int K = tl * FI;  // multiple of 4
      k_fillz<<<nblk((size_t)2 * 256 * 384), 256, 0, stream>>>((float*)Afz, (size_t)2 * 256 * 384);
      k_fillz<<<nblk((size_t)2 * 384 * 768), 256, 0, stream>>>((float*)Bfz, (size_t)2 * 384 * 768);
      k_fx<<<nblk((size_t)tl * Bn * tl * FI), 256, 0, stream>>>(bufB, din1, Afz, l, Bn, FI, NZ, N, LDA);
      k_fyb<<<nblk((size_t)tl * FI * tl * FO), 256, 0, stream>>>(khat1, dpi1, Bfz, l, FI, FO, NG, LDB);
      k_zgemm<<<(Mp / 16) * (Np / 16), 32, 0, stream>>>(Afz, Bfz, Cfz, Np / 16, K, LDA, LDB, LDC);
      k_contrib<<<nblk((size_t)Bn * FO * tl * tl), 256, 0, stream>>>(Cfz, dout1, bufA, l, Bn, FO, NZO, NO, LDC);
    }
    // inverse (unnormalized) DFT along gamma then alpha
    int Ri = Bn * FO * NZO * NO;  // 196608
    k_dft_t<32, 0, 0><<<Ri / 16, 32, 0, stream>>>(bufA, nullptr, bufB, W32i, 1, NO, 1);
    k_dft_t<32, 0, 0><<<Ri / 16, 32, 0, stream>>>(bufB, nullptr, bufA, W32i, NO, NO * NO, NO);
    size_t tot = (size_t)Bn * FO * NZO * NO * NO;
    k_final<<<nblk(tot), 256, 0, stream>>>(bufA, bias1, out, tot, NZO * NO * NO, FO);
  }

  // ============================ Layer 2 ============================
  // b_in=16 (n=32, nz=32), b_out=8 (no=16, nzo=16), fi=24, fo=32, ng=32
  {
    const int Bn = 8, FI = 24, FO = 32, N = 32, NZ = 32, NO = 16, NZO = 16, NG = 32;
    int R = Bn * FI * NZ * N;  // 196608
    // forward DFT along gamma: real input = relu(x1) from d_out
    k_dft_t<32, 1, 1><<<R / 16, 32, 0, stream>>>(nullptr, out, bufA, W32f, 1, N, 1);
    k_dft_t<32, 0, 0><<<R / 16, 32, 0, stream>>>(bufA, nullptr, bufB, W32f, N, N * N, N);
    size_t sflt = (size_t)2 * Bn * FO * NZO * NO * NO;
    k_fillz<<<nblk(sflt), 256, 0, stream>>>((float*)bufA, sflt);
    for (int l = 0; l < 8; ++l) {
      int tl = 2 * l + 1;
      int Mp = ((tl * Bn + 15) / 16) * 16;
      int Np = ((tl * FO + 15) / 16) * 16;
      int K = tl * FI;  // multiple of 4
      k_fillz<<<nblk((size_t)2 * 256 * 384), 256, 0, stream>>>((float*)Afz, (size_t)2 * 256 * 384);
      k_fillz<<<nblk((size_t)2 * 384 * 768), 256, 0, stream>>>((float*)Bfz, (size_t)2 * 384 * 768);
      k_fx<<<nblk((size_t)tl * Bn * tl * FI), 256, 0, stream>>>(bufB, din2, Afz, l, Bn, FI, NZ, N, LDA);
      k_fyb<<<nblk((size_t)tl * FI * tl * FO), 256, 0, stream>>>(khat2, dpi2t, Bfz, l, FI, FO, NG, LDB);
      k_zgemm<<<(Mp / 16) * (Np / 16), 32, 0, stream>>>(Afz, Bfz, Cfz, Np / 16, K, LDA, LDB, LDC);
      k_contrib<<<nblk((size_t)Bn * FO * tl * tl), 256, 0, stream>>>(Cfz, dout2, bufA, l, Bn, FO, NZO, NO, LDC);
    }
    int Ri = Bn * FO * NZO * NO;  // 32768
    k_dft_t<16, 0, 0><<<Ri / 16, 32, 0, stream>>>(bufA, nullptr, bufB, W16i, 1, NO, 1);
    k_dft_t<16, 0, 0><<<Ri / 16, 32, 0, stream>>>(bufB, nullptr, bufA, W16i, NO, NO * NO, NO);
    size_t tot = (size_t)Bn * FO * NZO * NO * NO;
    k_final<<<nblk(tot), 256, 0, stream>>>(bufA, bias2, out + 6291456, tot, NZO * NO * NO, FO);
  }
}